// elr_loss_21749714387538
// MI455X (gfx1250) — compile-verified
//
#include <hip/hip_runtime.h>
#include <hip/hip_bf16.h>

// ---------------------------------------------------------------------------
// ELR loss on MI455X (gfx1250).
// Dominant cost: 800MB read + 800MB write of the target buffer  -> ~69us at
// 23.3 TB/s. Bulk copy uses the CDNA5 async LDS-DMA path (no VGPR staging,
// 8 in-flight b128 transfers per wave, non-temporal hints since the stream is
// 4x the 192MB L2 and never re-read). Batch kernel (1024x1000) uses
// V_WMMA_F32_16X16X4_F32 with an all-ones B matrix to produce 16 row-sums per
// wave for the softmax-normalization (P = sum y) and the ELR dot-product term
// (Q = sum y^2), fused with the EMA row update.
// ---------------------------------------------------------------------------

typedef __attribute__((ext_vector_type(2))) float v2f;
typedef __attribute__((ext_vector_type(8))) float v8f;

#define C_DIM   1000
#define ROWS    16
#define LSTRIDE 1002          // even (8B-aligned b64 pair loads), low-conflict
#define B_DIM   1024
#define CLIP_LO 1e-4f
#define CLIP_HI (1.0f - 1e-4f)

#define CP_THREADS 256
#define CP_UNROLL  8

// ---------------------------------------------------------------------------
// Bulk copy: target -> d_out+1 via async LDS DMA (memory -> LDS -> memory).
// src chunks are 16B aligned; dst (d_out+1) is 4B aligned, which is legal for
// b128 global ops (dword alignment suffices in DWORD alignment mode).
// NT hints keep the 1.6GB stream from thrashing L2.
// ---------------------------------------------------------------------------
__global__ void __launch_bounds__(CP_THREADS)
elr_copy_async(const float* __restrict__ src, float* __restrict__ dst,
               long nchunks, long ntail)
{
    __shared__ char stage[CP_UNROLL][CP_THREADS * 16];
    const unsigned lds0 =
        (unsigned)(size_t)&stage[0][threadIdx.x * 16];   // low 32b == LDS offset

    const long stride = (long)gridDim.x * CP_THREADS * CP_UNROLL;
    for (long base = (long)blockIdx.x * CP_THREADS * CP_UNROLL;
         base < nchunks; base += stride) {
#pragma unroll
        for (int u = 0; u < CP_UNROLL; ++u) {
            long idx = base + (long)u * CP_THREADS + threadIdx.x;
            if (idx < nchunks) {
                const float* sp = src + idx * 4;
                unsigned lds = lds0 + u * (CP_THREADS * 16);
                asm volatile("global_load_async_to_lds_b128 %0, %1, off th:TH_LOAD_NT"
                             :: "v"(lds), "v"(sp) : "memory");
            }
        }
        asm volatile("s_wait_asynccnt 0x0" ::: "memory");  // loads landed in LDS
#pragma unroll
        for (int u = 0; u < CP_UNROLL; ++u) {
            long idx = base + (long)u * CP_THREADS + threadIdx.x;
            if (idx < nchunks) {
                float* dp = dst + idx * 4;
                unsigned lds = lds0 + u * (CP_THREADS * 16);
                asm volatile("global_store_async_from_lds_b128 %0, %1, off th:TH_STORE_NT"
                             :: "v"(dp), "v"(lds) : "memory");
            }
        }
        asm volatile("s_wait_asynccnt 0x0" ::: "memory");  // slots reusable
    }
    // scalar tail (n % 4 elements), normally zero for this problem size
    if (blockIdx.x == 0 && (long)threadIdx.x < ntail) {
        long i = nchunks * 4 + threadIdx.x;
        dst[i] = src[i];
    }
}

// ---------------------------------------------------------------------------
// Batch kernel: one wave per 16 rows (64 blocks x 32 threads).
// ---------------------------------------------------------------------------
__global__ void __launch_bounds__(32)
elr_batch(const float* __restrict__ output, const float* __restrict__ target,
          const int* __restrict__ label, const int* __restrict__ index,
          float* __restrict__ loss, float* __restrict__ newtgt)
{
    __shared__ float e_buf[ROWS * LSTRIDE];                 // exp(x - max)
    __shared__ float s_max[ROWS], s_sum[ROWS], s_P[ROWS], s_Q[ROWS];

    const int lane = threadIdx.x;
    const int row0 = blockIdx.x * ROWS;
    const int start = index[0] * B_DIM;

    // ---- pass 1: per-row max, exp, sum(exp); stash exp values in LDS ----
    for (int m = 0; m < ROWS; ++m) {
        const float* x = output + (size_t)(row0 + m) * C_DIM;
        float mx = -INFINITY;
        for (int j = lane; j < C_DIM; j += 32) mx = fmaxf(mx, x[j]);
        for (int o = 16; o; o >>= 1) mx = fmaxf(mx, __shfl_xor(mx, o, 32));
        float se = 0.0f;
        for (int j = lane; j < C_DIM; j += 32) {
            float e = __expf(x[j] - mx);
            e_buf[m * LSTRIDE + j] = e;
            se += e;
        }
        for (int o = 16; o; o >>= 1) se += __shfl_xor(se, o, 32);
        if (lane == 0) { s_max[m] = mx; s_sum[m] = se; }
    }
    __syncthreads();

    // ---- pass 2: WMMA row sums. A = 16x4 chunk of clipped y_pred (or y^2),
    //      B = ones(4x16)  =>  D[m][*] += sum_k A[m][k]. 250 K-chunks. ----
    const int   r    = lane & 15;            // A layout: lanes 0-15 -> M rows
    const int   cadd = (lane >> 4) * 2;      // upper half lanes hold K+2,K+3
    const float inv  = 1.0f / s_sum[r];
    v2f ones; ones.x = 1.0f; ones.y = 1.0f;
    v8f accP = {};
    v8f accQ = {};
    for (int k = 0; k < C_DIM / 4; ++k) {
        const int c0 = 4 * k + cadd;
        v2f e2 = *(const v2f*)&e_buf[r * LSTRIDE + c0];   // ds_load_b64
        v2f a, q;
        a.x = fminf(fmaxf(e2.x * inv, CLIP_LO), CLIP_HI);
        a.y = fminf(fmaxf(e2.y * inv, CLIP_LO), CLIP_HI);
        q.x = a.x * a.x;
        q.y = a.y * a.y;
        accP = __builtin_amdgcn_wmma_f32_16x16x4_f32(
                   false, a, false, ones, (short)0, accP, false, false);
        accQ = __builtin_amdgcn_wmma_f32_16x16x4_f32(
                   false, q, false, ones, (short)0, accQ, false, false);
    }
    // D layout: VGPR v, lanes 0-15 -> M=v ; lanes 16-31 -> M=v+8
    if (lane == 0)
        for (int v = 0; v < 8; ++v) { s_P[v] = accP[v];     s_Q[v] = accQ[v]; }
    if (lane == 16)
        for (int v = 0; v < 8; ++v) { s_P[v + 8] = accP[v]; s_Q[v + 8] = accQ[v]; }
    __syncthreads();

    // ---- pass 3: EMA row write (coalesced), R = sum(t*y), loss pieces ----
    for (int m = 0; m < ROWS; ++m) {
        const int row   = row0 + m;
        const float invm = 1.0f / s_sum[m];
        const float invP = 0.3f / s_P[m];        // 0.3 * y / P  and  0.3*Q/P
        const float* t = target + (size_t)(start + row) * C_DIM;
        float*       d = newtgt + (size_t)(start + row) * C_DIM;
        float R = 0.0f;
        for (int j = lane; j < C_DIM; j += 32) {
            float y  = fminf(fmaxf(e_buf[m * LSTRIDE + j] * invm, CLIP_LO), CLIP_HI);
            float tv = t[j];
            d[j] = 0.7f * tv + invP * y;
            R += tv * y;
        }
        for (int o = 16; o; o >>= 1) R += __shfl_xor(R, o, 32);
        if (lane == 0) {
            const float* x = output + (size_t)row * C_DIM;
            int lb   = label[row];
            float ce  = -(x[lb] - s_max[m] - __logf(s_sum[m]));
            float dot = 0.7f * R + invP * s_Q[m];
            float elr = __logf(1.0f - dot);
            atomicAdd(loss, (ce + 3.0f * elr) * (1.0f / (float)B_DIM));
        }
    }
}

__global__ void elr_zero_loss(float* loss)
{
    if (threadIdx.x == 0) *loss = 0.0f;
}

extern "C" void kernel_launch(void* const* d_in, const int* in_sizes, int n_in,
                              void* d_out, int out_size, void* d_ws, size_t ws_size,
                              hipStream_t stream)
{
    const float* output = (const float*)d_in[0];
    const float* target = (const float*)d_in[1];
    const int*   label  = (const int*)d_in[2];
    const int*   index  = (const int*)d_in[3];

    float* loss = (float*)d_out;
    float* newt = loss + 1;

    const long n       = (long)in_sizes[1];      // 200,000,000 floats
    const long nchunks = n / 4;
    const long ntail   = n & 3;

    elr_zero_loss<<<1, 32, 0, stream>>>(loss);
    elr_copy_async<<<4096, CP_THREADS, 0, stream>>>(target, newt, nchunks, ntail);
    elr_batch<<<B_DIM / ROWS, 32, 0, stream>>>(output, target, label, index,
                                               loss, newt);
}